// MPSLayer_53060025975088
// MI455X (gfx1250) — compile-verified
//
#include <hip/hip_runtime.h>
#include <stdint.h>

// ---------------------------------------------------------------------------
// MPS chain contraction: r_{t+1}[b,a] = sum_i r_t[b,i] * T_t[a,i,b]
// D = 128 (b,i,a), STEPS = 128.  Bandwidth bound: 1 GiB of T streamed once.
//
// Decomposition: 4 teams (32-wide b-blocks, lane<->b for coalescing),
// 32 WGs/team (4 a-rows each, 1 per wave).  Per-b independence means sync is
// team-local only.  T tiles are DMA'd global->LDS with the CDNA5 Tensor Data
// Mover (tensor_load_to_lds), double buffered with s_wait_tensorcnt.
// ---------------------------------------------------------------------------

#define DD 128
#define STEPS 128
#define BBLK 32                       // b per team (one 128B line of floats)
#define APW 4                         // a-rows per workgroup (1 per wave)
#define WGS_PER_TEAM (DD / APW)       // 32
#define NTEAMS (DD / BBLK)            // 4
#define NWG (NTEAMS * WGS_PER_TEAM)   // 128
#define BLOCK (APW * 32)              // 128 threads = 4 waves
#define NBAR (STEPS - 1)              // 127 inter-step team barriers

#define RBUF_FLOATS (NTEAMS * 2 * DD * BBLK)   // 32768 floats = 128 KB
#define FLAGS_OFF_BYTES (RBUF_FLOATS * 4)      // flags live after rbuf in d_ws

#define TILE_FLOATS (APW * DD * BBLK)          // 16384 floats = 64 KB per tile
#define LDS_R_FLOATS (DD * BBLK)               // 4096 floats  = 16 KB r stage
#define LDS_TOTAL_BYTES ((LDS_R_FLOATS + 2 * TILE_FLOATS) * 4)  // 144 KB

typedef unsigned int u32x4 __attribute__((ext_vector_type(4)));
typedef int i32x4 __attribute__((ext_vector_type(4)));
typedef int i32x8 __attribute__((ext_vector_type(8)));

// ---- TDM: DMA one [APW a][128 i][32 b] fp32 tile from global into LDS ------
// D# group0: count=1 | lds_addr | global_addr | type=2
// D# group1: data_size=4B, tile = 32 x 128 x APW, strides 128 / 16384 elems,
//            tensor dims set huge so the OOB clamp never triggers.
__device__ __forceinline__ void tdm_issue_tile(const float* gptr, uint32_t lds_byte_off) {
  uint64_t ga = (uint64_t)(uintptr_t)gptr;
  u32x4 g0 = { 1u,                                   // count=1, user descriptor
               lds_byte_off,                         // lds_addr (bytes)
               (uint32_t)ga,                         // global_addr[31:0]
               ((uint32_t)(ga >> 32) & 0x01FFFFFFu)  // global_addr[56:32]
                 | 0x80000000u };                    // type=2 ("image")
  i32x8 g1 = { (int)0x00020000,                      // data_size=2 (4 bytes)
               0,                                    // abar=0, dim0 lo16 (dim0=2^30)
               (int)0x00004000,                      // dim0 hi16, dim1 lo16
               (int)(0x00004000u | (32u << 16)),     // dim1 hi16 | tile_dim0=32 (b)
               (int)(128u | ((uint32_t)APW << 16)),  // tile_dim1=128 (i) | tile_dim2=APW (a)
               (int)128,                             // dim0_stride = 128 elems (i step)
               (int)0x40000000,                      // dim0_str hi16=0 | dim1_str lo16 (16384)
               0 };                                  // dim1_stride hi32 = 0
  i32x4 g2 = { (int)0x40000000, 1, 0, 0 };           // tensor_dim2 big, dim3=1, 3D tile
  i32x4 g3 = { 0, 0, 0, 0 };
#if defined(__clang_major__) && (__clang_major__ >= 23)
  i32x8 g4 = { 0, 0, 0, 0, 0, 0, 0, 0 };
  __builtin_amdgcn_tensor_load_to_lds(g0, g1, g2, g3, g4, 0);
#else
  __builtin_amdgcn_tensor_load_to_lds(g0, g1, g2, g3, 0);
#endif
}

__device__ __forceinline__ float ld_agent_f(const float* p) {
  return __hip_atomic_load(p, __ATOMIC_RELAXED, __HIP_MEMORY_SCOPE_AGENT);
}
__device__ __forceinline__ void st_agent_f(float* p, float v) {
  __hip_atomic_store(p, v, __ATOMIC_RELAXED, __HIP_MEMORY_SCOPE_AGENT);
}
__device__ __forceinline__ void st_flag(unsigned* p, unsigned v, int rel) {
  __hip_atomic_store(p, v, rel ? __ATOMIC_RELEASE : __ATOMIC_RELAXED,
                     __HIP_MEMORY_SCOPE_AGENT);
}
__device__ __forceinline__ unsigned ld_flag(const unsigned* p) {
  return __hip_atomic_load(p, __ATOMIC_ACQUIRE, __HIP_MEMORY_SCOPE_AGENT);
}
__device__ __forceinline__ void fence_rel_agent() {
  __builtin_amdgcn_fence(__ATOMIC_RELEASE, "agent");
}
__device__ __forceinline__ void fence_acq_agent() {
  __builtin_amdgcn_fence(__ATOMIC_ACQUIRE, "agent");
}

__global__ __launch_bounds__(BLOCK, 1) void mps_chain_kernel(
    const float* __restrict__ x, const float* __restrict__ Tg,
    float* __restrict__ out, float* __restrict__ rbuf,
    unsigned* __restrict__ flags) {
  extern __shared__ float smem[];   // [0,16K): r stage; [16K,80K): T0; [80K,144K): T1
  const int tid   = threadIdx.x;
  const int lane  = tid & 31;                     // lane <-> b within block
  const int wave  = tid >> 5;                     // wave <-> local a row
  const int team  = blockIdx.x / WGS_PER_TEAM;    // 32-wide b block
  const int wg    = blockIdx.x % WGS_PER_TEAM;    // 4-wide a block
  const int b0    = team * BBLK;
  const int abase = wg * APW;
  const int a     = abase + wave;

  float* ldsR = smem;
  unsigned* tflags = flags + (size_t)team * NBAR * WGS_PER_TEAM;

  // Launch-time cleanup of the two barrier slots the in-run resets can't cover.
  if (tid == 0) {
    st_flag(&tflags[0 * WGS_PER_TEAM + wg], 0u, 0);
    st_flag(&tflags[1 * WGS_PER_TEAM + wg], 0u, 0);
  }

  // Prologue: kick off the TDM for step 0 into buffer 0.
  if (tid == 0)
    tdm_issue_tile(Tg + (size_t)abase * (DD * DD) + b0,
                   (uint32_t)(LDS_R_FLOATS * 4));

  for (int t = 0; t < STEPS; ++t) {
    const int par = t & 1;

    // ---- stage r_t[b, i] for this team's 32 b's into LDS -------------------
    if (t == 0) {
      for (int k = tid; k < DD * BBLK; k += BLOCK) {
        const int i = k >> 5, b = k & 31;
        ldsR[k] = x[(size_t)(b0 + b) * DD + i];
      }
    } else {
      const float* rsrc = rbuf + (size_t)(team * 2 + ((t + 1) & 1)) * (DD * BBLK);
      for (int k = tid; k < DD * BBLK; k += BLOCK)
        ldsR[k] = ld_agent_f(rsrc + k);           // agent scope: bypass stale L0
    }

    // ---- prefetch next T tile via TDM, wait for current tile ---------------
    if (t + 1 < STEPS) {
      if (tid == 0)
        tdm_issue_tile(Tg + ((size_t)(t + 1) * DD + abase) * (DD * DD) + b0,
                       (uint32_t)((LDS_R_FLOATS + ((t + 1) & 1) * TILE_FLOATS) * 4));
      if (wave == 0) __builtin_amdgcn_s_wait_tensorcnt(1);  // current tile done
    } else {
      if (wave == 0) __builtin_amdgcn_s_wait_tensorcnt(0);
    }
    __syncthreads();

    // ---- compute: acc[b] = sum_i r[i,b] * T[a,i,b] (LDS, conflict-free) ----
    const float* Ta = smem + LDS_R_FLOATS + par * TILE_FLOATS
                    + wave * (DD * BBLK) + lane;
    const float* Ra = ldsR + lane;
    float acc = 0.0f;
#pragma unroll 16
    for (int i = 0; i < DD; ++i)
      acc = __builtin_fmaf(Ra[i * BBLK], Ta[i * BBLK], acc);

    if (t == STEPS - 1) {
      out[(size_t)(b0 + lane) * DD + a] = acc;    // out[b,a] = r_128[b,a]
    } else {
      // publish new r row, then team barrier (per-WG slot flags in scratch)
      st_agent_f(rbuf + ((size_t)(team * 2 + par) * DD + a) * BBLK + lane, acc);
      fence_rel_agent();
      __syncthreads();
      if (tid == 0) {
        if (t + 2 < NBAR)                         // self-clean slot used at t+2
          st_flag(&tflags[(t + 2) * WGS_PER_TEAM + wg], 0u, 0);
        st_flag(&tflags[t * WGS_PER_TEAM + wg], 1u, 1);
        for (int w = 0; w < WGS_PER_TEAM; ++w)
          while (ld_flag(&tflags[t * WGS_PER_TEAM + w]) != 1u)
            __builtin_amdgcn_s_sleep(1);
        fence_acq_agent();
      }
      __syncthreads();
    }
  }

  // Leave slots 0/1 clean for the next (graph-replayed) launch.
  if (tid == 0) {
    st_flag(&tflags[0 * WGS_PER_TEAM + wg], 0u, 0);
    st_flag(&tflags[1 * WGS_PER_TEAM + wg], 0u, 0);
  }
}

extern "C" void kernel_launch(void* const* d_in, const int* in_sizes, int n_in,
                              void* d_out, int out_size, void* d_ws, size_t ws_size,
                              hipStream_t stream) {
  (void)in_sizes; (void)n_in; (void)out_size; (void)ws_size;
  const float* x  = (const float*)d_in[0];       // (128,128)  r_0[b,i]
  const float* Tg = (const float*)d_in[1];       // (128,128,128,128) [t][a][i][b]
  float* out      = (float*)d_out;               // (128,128)  out[b,a]
  float* rbuf     = (float*)d_ws;                // team ping-pong r state
  unsigned* flags = (unsigned*)((char*)d_ws + FLAGS_OFF_BYTES);
  mps_chain_kernel<<<dim3(NWG), dim3(BLOCK), LDS_TOTAL_BYTES, stream>>>(
      x, Tg, out, rbuf, flags);
}